// DepthwiseConv_15642270892748
// MI455X (gfx1250) — compile-verified
//
#include <hip/hip_runtime.h>
#include <cstdint>

// Depthwise 3x3 conv, NHWC, SAME padding, stride 1, fp32 + bias.
// B=32, H=W=112, C=192. Memory-bound (AI ~2.25 FLOP/B): optimize the data
// path with CDNA5 async global->LDS copies and NT stores, not WMMA.

typedef float v4f __attribute__((ext_vector_type(4)));

#define BATCH 32
#define HH 112
#define WW 112
#define CC 192
#define C4 48            // CC / 4
#define WT 16            // output W-tile per block
#define HT 14            // output H-rows per block
#define NTHR 192         // threads per block (6 waves of 32)
#define SLOT_V4 (18 * C4) // one staged input row: 18 px (16 + 2 halo) * 48 v4f

// Async copy of 16 bytes/lane from global (contiguous) into LDS.
// gfx1250: each lane stores the 16 B read at saddr + its vaddr offset to LDS at its vdst address
__device__ __forceinline__ void async_row_copy(unsigned lds0,
                                               const float* __restrict__ g,
                                               int chunks, int tid) {
  unsigned long long gaddr = (unsigned long long)(uintptr_t)g;
  for (int i = tid; i < chunks; i += NTHR) {
    unsigned off = (unsigned)i * 16u;
    asm volatile("global_load_async_to_lds_b128 %0, %1, %2"
                 :: "v"(lds0 + off), "v"(off), "s"(gaddr)
                 : "memory");
  }
}

// Stage one input row (global row hg) into ring slot, or zero-fill if OOB.
__device__ __forceinline__ void load_row(const float* __restrict__ xb, int hg,
                                         int slot, int wstart, int ldsx0,
                                         int chunks, v4f* smem, int tid) {
  if (hg < 0 || hg >= HH) {
    v4f z = {0.f, 0.f, 0.f, 0.f};
    for (int i = tid; i < SLOT_V4; i += NTHR) smem[slot * SLOT_V4 + i] = z;
  } else {
    const float* g = xb + ((size_t)hg * WW + wstart) * CC;
    // low 32 bits of a generic pointer into __shared__ == LDS byte address
    unsigned lds0 = (unsigned)(uintptr_t)(&smem[slot * SLOT_V4 + ldsx0 * C4]);
    async_row_copy(lds0, g, chunks, tid);
  }
}

__global__ void __launch_bounds__(NTHR)
dwconv3x3_kernel(const float* __restrict__ x, const float* __restrict__ w,
                 const float* __restrict__ bias, float* __restrict__ out) {
  __shared__ v4f smem[4 * SLOT_V4];   // 55,296 B: 4-deep row ring buffer

  const int tid = threadIdx.x;
  int bx = blockIdx.x;
  const int wx = bx % 7;  bx /= 7;    // W tile index (7 * 16 = 112)
  const int hc = bx % 8;  bx /= 8;    // H chunk index (8 * 14 = 112)
  const int b  = bx;                  // batch
  const int w0 = wx * WT;
  const int h0 = hc * HT;

  const int c4  = tid % C4;           // float4 channel group
  const int px0 = (tid / C4) * 4;     // first of 4 adjacent output pixels

  // Weights (HW1C layout -> [k][c]) and bias held in registers for the chunk.
  const v4f* wv = (const v4f*)w;
  v4f wk[9];
#pragma unroll
  for (int k = 0; k < 9; ++k) wk[k] = wv[k * C4 + c4];
  const v4f bv = ((const v4f*)bias)[c4];

  // W-tile staging geometry (clip halo at image edges).
  const int wstart = (wx == 0) ? 0 : (w0 - 1);
  const int ldsx0  = (wx == 0) ? 1 : 0;
  const int npx    = 18 - (wx == 0) - (wx == 6);
  const int chunks = npx * C4;        // 16B chunks per staged row

  const float* xb = x   + (size_t)b * HH * WW * CC;
  float*       ob = out + (size_t)b * HH * WW * CC;

  // Edge tiles: zero the missing halo column once in every ring slot
  // (async loads never touch it afterwards).
  if (wx == 0 || wx == 6) {
    const int hx = (wx == 0) ? 0 : 17;
    v4f z = {0.f, 0.f, 0.f, 0.f};
    for (int i = tid; i < 4 * C4; i += NTHR)
      smem[(i / C4) * SLOT_V4 + hx * C4 + (i % C4)] = z;
  }

  // Prologue: stage rows h0-1, h0, h0+1. slot(hg) = (hg+1) & 3.
  load_row(xb, h0 - 1, (h0    ) & 3, wstart, ldsx0, chunks, smem, tid);
  load_row(xb, h0    , (h0 + 1) & 3, wstart, ldsx0, chunks, smem, tid);
  load_row(xb, h0 + 1, (h0 + 2) & 3, wstart, ldsx0, chunks, smem, tid);
  asm volatile("s_wait_asynccnt 0" ::: "memory");
  __syncthreads();

  for (int it = 0; it < HT; ++it) {
    const int h  = h0 + it;
    const int hn = h + 2;
    // Prefetch row h+2 into the free ring slot; overlaps with compute below.
    if (hn <= h0 + HT)
      load_row(xb, hn, (hn + 1) & 3, wstart, ldsx0, chunks, smem, tid);

    // Slots holding rows h-1, h, h+1.
    const int sA = ((h    ) & 3) * SLOT_V4;
    const int sB = ((h + 1) & 3) * SLOT_V4;
    const int sC = ((h + 2) & 3) * SLOT_V4;

    v4f a0 = bv, a1 = bv, a2 = bv, a3 = bv;
#pragma unroll
    for (int r = 0; r < 3; ++r) {
      const int sb = (r == 0) ? sA : ((r == 1) ? sB : sC);
      const int tb = sb + px0 * C4 + c4;
      // 6 taps cover 4 adjacent outputs (register reuse in W).
      const v4f t0 = smem[tb];
      const v4f t1 = smem[tb + 1 * C4];
      const v4f t2 = smem[tb + 2 * C4];
      const v4f t3 = smem[tb + 3 * C4];
      const v4f t4 = smem[tb + 4 * C4];
      const v4f t5 = smem[tb + 5 * C4];
      const v4f k0 = wk[r * 3 + 0], k1 = wk[r * 3 + 1], k2 = wk[r * 3 + 2];
      a0 += k0 * t0 + k1 * t1 + k2 * t2;
      a1 += k0 * t1 + k1 * t2 + k2 * t3;
      a2 += k0 * t2 + k1 * t3 + k2 * t4;
      a3 += k0 * t3 + k1 * t4 + k2 * t5;
    }

    // Non-temporal 128-bit stores: output is never re-read, keep L2 for input.
    v4f* orow = (v4f*)(ob + ((size_t)h * WW + (w0 + px0)) * CC) + c4;
    __builtin_nontemporal_store(a0, orow);
    __builtin_nontemporal_store(a1, orow + 1 * C4);
    __builtin_nontemporal_store(a2, orow + 2 * C4);
    __builtin_nontemporal_store(a3, orow + 3 * C4);

    asm volatile("s_wait_asynccnt 0" ::: "memory");
    __syncthreads();
  }
}

extern "C" void kernel_launch(void* const* d_in, const int* in_sizes, int n_in,
                              void* d_out, int out_size, void* d_ws, size_t ws_size,
                              hipStream_t stream) {
  const float* x  = (const float*)d_in[0];   // (32,112,112,192) f32
  const float* w  = (const float*)d_in[1];   // (3,3,1,192) f32
  const float* bb = (const float*)d_in[2];   // (192,) f32
  float* out = (float*)d_out;                // (32,112,112,192) f32

  dim3 grid(7 * 8 * BATCH);                  // 1792 blocks
  dim3 block(NTHR);                          // 192 threads = 6 waves
  dwconv3x3_kernel<<<grid, block, 0, stream>>>(x, w, bb, out);
}